// Splatter_3736621547578
// MI455X (gfx1250) — compile-verified
//
#include <hip/hip_runtime.h>

#define GRID_G   106
#define KR       2
#define NPTS     131072
#define BATCH    2
#define CROP     100
#define CROPTOT  (CROP*CROP*CROP)        // 1,000,000 cells per batch
#define OUT_PER  (BATCH*CROPTOT)         // 2,000,000 floats per output array
#define EPSV     1e-9f

// ---------------------------------------------------------------------------
// Kernel 1: zero the output buffer (grid + acc halves), vectorized b128 stores
// ---------------------------------------------------------------------------
__global__ void zero_kernel(float4* __restrict__ out, int n4) {
    int i = blockIdx.x * blockDim.x + threadIdx.x;
    if (i < n4) out[i] = make_float4(0.f, 0.f, 0.f, 0.f);
}

// ---------------------------------------------------------------------------
// Kernel 2: splat. One thread per point. Async-stage point data into LDS,
// compute separable triangle weights, scatter-add into cropped grid/acc via
// native f32 global atomics (L2-resident footprint -> atomic-rate bound).
// ---------------------------------------------------------------------------
__global__ __launch_bounds__(256) void splat_kernel(
        const float* __restrict__ points,     // (B, N, 3)
        const float* __restrict__ densities,  // (B, N)
        const float* __restrict__ bbox,       // (1, 3, 2) = [xlo,xhi,ylo,yhi,zlo,zhi]
        float* __restrict__ out)              // [grid: OUT_PER][acc: OUT_PER]
{
    __shared__ float sp[256 * 4];             // per-lane {x,y,z,density}

    const int lane = threadIdx.x;
    const int tid  = blockIdx.x * 256 + lane; // < BATCH*NPTS exactly
    const int b    = tid / NPTS;

    const float* pptr = points + (size_t)tid * 3;
    const float* dptr = densities + tid;

    // ---- async global -> LDS staging (CDNA5 ASYNCcnt pipe) ----
    unsigned l0 = (unsigned)(size_t)&sp[lane * 4];   // LDS byte offset of this lane's slot
    asm volatile("global_load_async_to_lds_b64 %0, %1, off"
                 :: "v"(l0),       "v"(pptr)     : "memory");
    asm volatile("global_load_async_to_lds_b32 %0, %1, off"
                 :: "v"(l0 + 8u),  "v"(pptr + 2) : "memory");
    asm volatile("global_load_async_to_lds_b32 %0, %1, off"
                 :: "v"(l0 + 12u), "v"(dptr)     : "memory");

    // ---- uniform bbox math overlapped with the async transfer ----
    // Reference: pad = (KR+1)*CELL = 0.03; pg = (p - lo - CELL/2)/(hi-lo)*G
    // Grid dim order is reversed coords: dim0 = z, dim1 = y, dim2 = x.
    const float pad = 0.03f;
    float lo2 = bbox[4] - pad, hi2 = bbox[5] + pad;  // dim0 <- z
    float lo1 = bbox[2] - pad, hi1 = bbox[3] + pad;  // dim1 <- y
    float lo0 = bbox[0] - pad, hi0 = bbox[1] + pad;  // dim2 <- x
    float s0g = (float)GRID_G / (hi2 - lo2);
    float s1g = (float)GRID_G / (hi1 - lo1);
    float s2g = (float)GRID_G / (hi0 - lo0);

    asm volatile("s_wait_asynccnt 0" ::: "memory");
    // Each lane reads only its own staged slot -> no workgroup barrier needed.
    float px  = sp[lane * 4 + 0];
    float py  = sp[lane * 4 + 1];
    float pz  = sp[lane * 4 + 2];
    float den = sp[lane * 4 + 3];

    // pg per grid dim (dim0=z, dim1=y, dim2=x)
    float pg0 = (pz - lo2 - 0.005f) * s0g;
    float pg1 = (py - lo1 - 0.005f) * s1g;
    float pg2 = (px - lo0 - 0.005f) * s2g;
    int   pgi0 = (int)rintf(pg0);
    int   pgi1 = (int)rintf(pg1);
    int   pgi2 = (int)rintf(pg2);

    // separable triangle weights, off in [-2,2]: w = max(1 - |pgi+off-pg|/2, 0)
    float w0[5], w1[5], w2[5];
    #pragma unroll
    for (int o = 0; o < 5; ++o) {
        w0[o] = fmaxf(1.f - 0.5f * fabsf((float)(pgi0 + o - KR) - pg0), 0.f);
        w1[o] = fmaxf(1.f - 0.5f * fabsf((float)(pgi1 + o - KR) - pg1), 0.f);
        w2[o] = fmaxf(1.f - 0.5f * fabsf((float)(pgi2 + o - KR) - pg2), 0.f);
    }

    const int bOff = b * CROPTOT;
    float* __restrict__ gout = out;
    float* __restrict__ aout = out + OUT_PER;

    #pragma unroll
    for (int i = 0; i < 5; ++i) {
        int o0 = pgi0 + i - (KR + 3);                 // cropped z index
        float wz = w0[i];
        if (wz == 0.f || (unsigned)o0 >= (unsigned)CROP) continue;
        #pragma unroll
        for (int j = 0; j < 5; ++j) {
            int o1 = pgi1 + j - (KR + 3);             // cropped y index
            float wzy = wz * w1[j];
            if (wzy == 0.f || (unsigned)o1 >= (unsigned)CROP) continue;
            int base = bOff + o0 * (CROP * CROP) + o1 * CROP;
            #pragma unroll
            for (int k = 0; k < 5; ++k) {
                int o2 = pgi2 + k - (KR + 3);         // cropped x index
                float w = wzy * w2[k];
                if (w == 0.f || (unsigned)o2 >= (unsigned)CROP) continue;
                atomicAdd(gout + base + o2, w * den); // grid += w * density
                atomicAdd(aout + base + o2, w);       // acc  += w
            }
        }
    }
}

// ---------------------------------------------------------------------------
// Kernel 3: normalize grid in place: grid = acc>0 ? grid/(acc+eps) : grid
// (acc half of d_out already holds the final acc output)
// ---------------------------------------------------------------------------
__global__ void norm_kernel(float* __restrict__ out) {
    int i = blockIdx.x * blockDim.x + threadIdx.x;
    if (i < OUT_PER) {
        float a = out[OUT_PER + i];
        if (a > 0.f) out[i] = out[i] / (a + EPSV);
    }
}

// ---------------------------------------------------------------------------
extern "C" void kernel_launch(void* const* d_in, const int* in_sizes, int n_in,
                              void* d_out, int out_size, void* d_ws, size_t ws_size,
                              hipStream_t stream) {
    const float* points    = (const float*)d_in[0];
    const float* densities = (const float*)d_in[1];
    const float* bbox      = (const float*)d_in[2];
    float* out = (float*)d_out;

    int n4 = out_size / 4;                              // 1,000,000 float4s
    zero_kernel<<<(n4 + 255) / 256, 256, 0, stream>>>((float4*)out, n4);

    int nthreads = BATCH * NPTS;                        // 262,144
    splat_kernel<<<nthreads / 256, 256, 0, stream>>>(points, densities, bbox, out);

    norm_kernel<<<(OUT_PER + 255) / 256, 256, 0, stream>>>(out);
}